// SimpleRNNDecoder_23819888624095
// MI455X (gfx1250) — compile-verified
//
#include <hip/hip_runtime.h>
#include <hip/hip_bf16.h>

typedef float v2f __attribute__((ext_vector_type(2)));
typedef float v8f __attribute__((ext_vector_type(8)));

#define BATCH  32
#define NNODES 128
#define NIN    128
#define NHID   256
#define NOUT   128
#define MROWS  (BATCH * NNODES)   // 4096 rows of x / PQ / T / out
#define HT     64                 // hidden-channel tile for k2 LDS staging

// ---------------------------------------------------------------------------
// Kernel 1: PQ = x @ [W1_top | W1_bot]  (M=4096, K=128, N=512), +b1 on Q half
//   PQ[:,   0:256] = x @ W1[0:128, :]          (P)
//   PQ[:, 256:512] = x @ W1[128:256, :] + b1   (Q)
// One wave per 16x16 tile, f32 WMMA, K stepped by 4.
// ---------------------------------------------------------------------------
__global__ __launch_bounds__(128) void k1_xw1(const float* __restrict__ X,
                                              const float* __restrict__ W1,
                                              const float* __restrict__ b1,
                                              float* __restrict__ PQ) {
  const int lane = threadIdx.x & 31;
  const int wave = threadIdx.x >> 5;
  const int lo   = lane & 15;       // M row (A) / N col (B,C)
  const int hi   = lane >> 4;       // K sub-pair select (A,B) / M half (C)
  const int m0   = blockIdx.x * 16;
  const int n0   = (blockIdx.y * 4 + wave) * 16;

  // W1' [128 x 512]: col n<256 -> W1[k,n]; col n>=256 -> W1[128+k, n-256].
  //   (128+k)*NHID + (n-NHID) == k*NHID + n + (128*NHID - NHID)
  const float* __restrict__ Wb = W1 + ((n0 < NHID) ? 0 : (128 * NHID - NHID));
  const int col = n0 + lo;

  v8f acc = {};
  for (int kk = 0; kk < NIN; kk += 4) {
    const int ka = kk + 2 * hi;
    v2f a, b;
    a.x = X[(m0 + lo) * NIN + ka];
    a.y = X[(m0 + lo) * NIN + ka + 1];
    b.x = Wb[ka * NHID + col];
    b.y = Wb[(ka + 1) * NHID + col];
    acc = __builtin_amdgcn_wmma_f32_16x16x4_f32(
        false, a, false, b, (short)0, acc, false, false);
  }

  const float bias = (n0 >= NHID) ? b1[col - NHID] : 0.0f;
  #pragma unroll
  for (int r = 0; r < 8; ++r) {
    const int row = m0 + r + 8 * hi;
    PQ[(size_t)row * 512 + col] = acc[r] + bias;
  }
}

// ---------------------------------------------------------------------------
// Kernel 2: sender reduction with LDS-staged P tile.
//   T[b*128+j, h] = sum_{i != j} relu(P[b,i,h] + Q[b,j,h])
// Block = (b, h-tile of 64 channels); P[b, :, h0:h0+64] (32 KB) lives in LDS
// and is reused by all 128 receivers j -> ~64x less global traffic than the
// naive per-(b,j) formulation. Sum over all i, then subtract the i==j term.
// ---------------------------------------------------------------------------
__global__ __launch_bounds__(256) void k2_reduce(const float* __restrict__ PQ,
                                                 float* __restrict__ T) {
  __shared__ float Pl[NNODES * HT];           // 128 x 64 f32 = 32 KB
  const int b  = blockIdx.x;
  const int h0 = blockIdx.y * HT;
  const int t  = threadIdx.x;

  // stage P[b, :, h0:h0+HT] into LDS (coalesced, 64-wide rows)
  #pragma unroll
  for (int c = 0; c < (NNODES * HT) / 256; ++c) {
    const int idx = c * 256 + t;
    const int i   = idx >> 6;                 // / HT
    const int hh  = idx & (HT - 1);
    Pl[idx] = PQ[(size_t)(b * NNODES + i) * 512 + h0 + hh];
  }
  __syncthreads();

  const int hh = t & (HT - 1);                // channel within tile
  const int jg = t >> 6;                      // receiver group 0..3
  for (int j = jg; j < NNODES; j += 4) {
    const float qv = PQ[(size_t)(b * NNODES + j) * 512 + NHID + h0 + hh];
    float acc = 0.0f;
    #pragma unroll 8
    for (int i = 0; i < NNODES; ++i) {
      const float v = Pl[i * HT + hh] + qv;
      acc += (v > 0.0f) ? v : 0.0f;
    }
    // remove the (i == j) self-edge contribution
    const float vj = Pl[j * HT + hh] + qv;
    acc -= (vj > 0.0f) ? vj : 0.0f;

    T[(size_t)(b * NNODES + j) * NHID + h0 + hh] = acc;
  }
}

// ---------------------------------------------------------------------------
// Kernel 3: out = (T @ W2 + 127*b2) / (127 + 1e-6)   (M=4096, K=256, N=128)
// ---------------------------------------------------------------------------
__global__ __launch_bounds__(128) void k3_tw2(const float* __restrict__ T,
                                              const float* __restrict__ W2,
                                              const float* __restrict__ b2,
                                              float* __restrict__ OUT) {
  const int lane = threadIdx.x & 31;
  const int wave = threadIdx.x >> 5;
  const int lo   = lane & 15;
  const int hi   = lane >> 4;
  const int m0   = blockIdx.x * 16;
  const int n0   = (blockIdx.y * 4 + wave) * 16;
  const int col  = n0 + lo;

  v8f acc = {};
  for (int kk = 0; kk < NHID; kk += 4) {
    const int ka = kk + 2 * hi;
    v2f a, b;
    a.x = T[(size_t)(m0 + lo) * NHID + ka];
    a.y = T[(size_t)(m0 + lo) * NHID + ka + 1];
    b.x = W2[ka * NOUT + col];
    b.y = W2[(ka + 1) * NOUT + col];
    acc = __builtin_amdgcn_wmma_f32_16x16x4_f32(
        false, a, false, b, (short)0, acc, false, false);
  }

  const float invn = 1.0f / (127.0f + 1e-6f);
  const float bias = 127.0f * b2[col];
  #pragma unroll
  for (int r = 0; r < 8; ++r) {
    const int row = m0 + r + 8 * hi;
    OUT[(size_t)row * NOUT + col] = (acc[r] + bias) * invn;
  }
}

// ---------------------------------------------------------------------------
// inputs (setup_inputs order): 0:x 1:rel_type 2:rel_rec 3:rel_send
//                              4:W1 5:b1 6:W2 7:b2   (all float32)
// output: [32,128,128] float32
// workspace: PQ (4096*512 f32 = 8 MB) | T (4096*256 f32 = 4 MB)
// ---------------------------------------------------------------------------
extern "C" void kernel_launch(void* const* d_in, const int* in_sizes, int n_in,
                              void* d_out, int out_size, void* d_ws, size_t ws_size,
                              hipStream_t stream) {
  (void)in_sizes; (void)n_in; (void)out_size; (void)ws_size;

  const float* x  = (const float*)d_in[0];
  const float* W1 = (const float*)d_in[4];
  const float* b1 = (const float*)d_in[5];
  const float* W2 = (const float*)d_in[6];
  const float* b2 = (const float*)d_in[7];
  float* out = (float*)d_out;

  float* PQ = (float*)d_ws;                       // [4096, 512]
  float* T  = PQ + (size_t)MROWS * 512;           // [4096, 256]

  // k1: 256 M-tiles x 8 blocks of 4 N-tiles (N=512)
  k1_xw1<<<dim3(MROWS / 16, (2 * NHID / 16) / 4), 128, 0, stream>>>(x, W1, b1, PQ);
  // k2: one block per (batch, 64-channel tile); P tile staged in LDS
  k2_reduce<<<dim3(BATCH, NHID / HT), 256, 0, stream>>>(PQ, T);
  // k3: 256 M-tiles x 2 blocks of 4 N-tiles (N=128)
  k3_tw2<<<dim3(MROWS / 16, (NOUT / 16) / 4), 128, 0, stream>>>(T, W2, b2, out);
}